// FlowExtract3D_83769042141670
// MI455X (gfx1250) — compile-verified
//
#include <hip/hip_runtime.h>
#include <math.h>

typedef float v2f __attribute__((ext_vector_type(2)));
typedef float v8f __attribute__((ext_vector_type(8)));

#define CCH 64
#define S 40
#define MD 6
#define NVOX (S * S * S)      // 64000
#define INV_TEMP 1.0e4f

// ---------------- Kernel 1: reciprocal L2 norms over channel dim ----------------
__global__ void rnorm_kernel(const float* __restrict__ in1,
                             const float* __restrict__ in2,
                             float* __restrict__ rn1,
                             float* __restrict__ rn2) {
    int v = blockIdx.x * blockDim.x + threadIdx.x;
    if (v >= NVOX) return;
    float s1 = 0.f, s2 = 0.f;
    for (int c = 0; c < CCH; ++c) {
        float a = in1[c * NVOX + v];
        float b = in2[c * NVOX + v];
        s1 += a * a;
        s2 += b * b;
    }
    rn1[v] = 1.0f / fmaxf(sqrtf(s1), 1e-12f);
    rn2[v] = 1.0f / fmaxf(sqrtf(s2), 1e-12f);
}

// ---------------- Kernel 2: banded-WMMA correlation + fused online softmax ----------------
// grid = (1600 = x*40+y, 13 = dy+6), block = 64 threads (2 waves)
// LDS is K-pair interleaved: X2[kp][col][j] holds element (K=2*kp+j, col) so each
// WMMA fragment is one aligned ds_load_b64 straight into an even VGPR pair.
__launch_bounds__(64)
__global__ void corrflow_kernel(const float* __restrict__ in1,
                                const float* __restrict__ in2,
                                const float* __restrict__ rn1,
                                const float* __restrict__ rn2,
                                float* __restrict__ out) {
    __shared__ float A2[32 * 48 * 2];   // [kp][z(40->48 pad)][2]  : in1 line
    __shared__ float B2[32 * 64 * 2];   // [kp][zp(52->64 pad)][2] : padded in2 line
    __shared__ float simb[S * 13];      // band buffer: [z][dz+6]

    const int tid  = threadIdx.x;
    const int lane = tid & 31;
    const int wv   = tid >> 5;          // wave id = N-tile id (0/1)
    const int half = lane >> 4;         // selects K pair index parity
    const int l    = lane & 15;

    const int xy = blockIdx.x;
    const int x  = xy / S;
    const int y  = xy - x * S;
    const int dy = (int)blockIdx.y - MD;
    const int yy = y + dy;
    const int base1 = x * S * S + y * S;
    const float dyf = (float)dy;

    // Stage A once: normalized in1 line, zero rows z>=40
    for (int idx = tid; idx < 64 * 48; idx += 64) {
        int c = idx / 48;
        int z = idx - c * 48;
        float v = 0.f;
        if (z < S) v = in1[c * NVOX + base1 + z] * rn1[base1 + z];
        A2[(((c >> 1) * 48) + z) * 2 + (c & 1)] = v;
    }

    // Online softmax state, one voxel per thread (threads 0..39)
    float m = -1.0e30f, ssum = 0.f, fx = 0.f, fy = 0.f, fz = 0.f;

    for (int dxi = 0; dxi < 13; ++dxi) {
        const int dx = dxi - MD;
        const int xx = x + dx;
        __syncthreads();   // prior iter's simb/Bld consumers done (also covers A staging)

        // Stage B: normalized, zero-padded in2 line at (xx, yy); col zp maps to z' = zp-6
        {
            const bool lineok = (xx >= 0) && (xx < S) && (yy >= 0) && (yy < S);
            const int base2 = lineok ? (xx * S * S + yy * S) : 0;
            for (int idx = tid; idx < 64 * 64; idx += 64) {
                int c  = idx >> 6;
                int zp = idx & 63;
                int z2 = zp - MD;
                float v = 0.f;
                if (lineok && z2 >= 0 && z2 < S)
                    v = in2[c * NVOX + base2 + z2] * rn2[base2 + z2];
                B2[(((c >> 1) * 64) + zp) * 2 + (c & 1)] = v;
            }
        }
        __syncthreads();

        // Banded GEMM: D[z][zp] = sum_c A[c][z] * B[c][zp]; 3 M-tiles, this wave's N-tile
        v8f acc0 = {}, acc1 = {}, acc2 = {};
        #pragma unroll 4
        for (int k = 0; k < 16; ++k) {
            const int kp = 2 * k + half;            // K pair index (kb = 2*kp)
            const v2f* arow = (const v2f*)&A2[(kp * 48 + l) * 2];
            const v2f* brow = (const v2f*)&B2[(kp * 64 + wv * 16 + l) * 2];
            v2f a, b;
            a = arow[0];  b = brow[0];              // M-tile 0, N base 0  + wv*16
            acc0 = __builtin_amdgcn_wmma_f32_16x16x4_f32(false, a, false, b,
                                                         (short)0, acc0, false, false);
            a = arow[16]; b = brow[16];             // M-tile 1, N base 16 + wv*16
            acc1 = __builtin_amdgcn_wmma_f32_16x16x4_f32(false, a, false, b,
                                                         (short)0, acc1, false, false);
            a = arow[32]; b = brow[32];             // M-tile 2, N base 32 + wv*16
            acc2 = __builtin_amdgcn_wmma_f32_16x16x4_f32(false, a, false, b,
                                                         (short)0, acc2, false, false);
        }

        // Scatter band entries: z = mt*16 + v + 8*half; zp = mt*16 + wv*16 + l; dz = zp-6-z
        #pragma unroll
        for (int mt = 0; mt < 3; ++mt) {
            v8f a = (mt == 0) ? acc0 : ((mt == 1) ? acc1 : acc2);
            #pragma unroll
            for (int v = 0; v < 8; ++v) {
                int z  = mt * 16 + v + 8 * half;
                int dz = wv * 16 + l - v - 8 * half - MD;   // mt cancels
                if (z < S && dz >= -MD && dz <= MD)
                    simb[z * 13 + dz + MD] = a[v];
            }
        }
        __syncthreads();

        // Branchless online softmax update over 13 dz values for this (dx, dy)
        if (tid < S) {
            const float dxf = (float)dx;
            #pragma unroll
            for (int i = 0; i < 13; ++i) {
                float v    = simb[tid * 13 + i];
                float dzf  = (float)(i - MD);
                float newm = fmaxf(m, v);
                float sc   = __expf((m - newm) * INV_TEMP);   // 1 if m stays max
                float e    = __expf((v - newm) * INV_TEMP);
                ssum = ssum * sc + e;
                fx   = fx * sc + e * dxf;
                fy   = fy * sc + e * dyf;
                fz   = fz * sc + e * dzf;
                m = newm;
            }
        }
    }

    // flow = weighted offsets / sum of weights; out layout [3][40][40][40]
    if (tid < S) {
        float inv = 1.0f / ssum;
        int o = base1 + tid;
        out[0 * NVOX + o] = fx * inv;
        out[1 * NVOX + o] = fy * inv;
        out[2 * NVOX + o] = fz * inv;
    }
}

extern "C" void kernel_launch(void* const* d_in, const int* in_sizes, int n_in,
                              void* d_out, int out_size, void* d_ws, size_t ws_size,
                              hipStream_t stream) {
    (void)in_sizes; (void)n_in; (void)out_size; (void)ws_size;
    const float* in1 = (const float*)d_in[0];
    const float* in2 = (const float*)d_in[1];
    float* out = (float*)d_out;
    float* rn1 = (float*)d_ws;          // 64000 floats
    float* rn2 = rn1 + NVOX;            // 64000 floats (512 KB total)

    rnorm_kernel<<<(NVOX + 255) / 256, 256, 0, stream>>>(in1, in2, rn1, rn2);

    dim3 grid(S * S, 13);
    corrflow_kernel<<<grid, 64, 0, stream>>>(in1, in2, rn1, rn2, out);
}